// MultiHeadGeometricRelativeAttention_32014686225111
// MI455X (gfx1250) — compile-verified
//
#include <hip/hip_runtime.h>
#include <hip/hip_bf16.h>

typedef __attribute__((ext_vector_type(2))) float v2f;
typedef __attribute__((ext_vector_type(8))) float v8f;

#define HEADS 8
#define DHEAD 64
#define SEQN  2048
#define BATCH 2
#define DIM   512
#define INNER 512
#define QKVW  1536
#define TOKENS (BATCH * SEQN)

#define WMMA_F32(a, b, c) \
    __builtin_amdgcn_wmma_f32_16x16x4_f32(false, (a), false, (b), (short)0, (c), false, false)

// ---------------------------------------------------------------------------
// fp32 GEMM with bias: C[M,N] = A[M,K] @ B[K,N] + bias[N]
// One wave computes a 16x64 strip (4 WMMA tiles). Fragment loads are
// software-pipelined one k-step ahead with a peeled epilogue so the steady-
// state loop has unconditional loads (no guards / zero-fills).
// ---------------------------------------------------------------------------
__global__ __launch_bounds__(32) void gemm_bias_wmma(const float* __restrict__ A,
                                                     const float* __restrict__ Bm,
                                                     const float* __restrict__ bias,
                                                     float* __restrict__ C,
                                                     int M, int N, int K) {
    const int n0   = blockIdx.x * 64;
    const int m0   = blockIdx.y * 16;
    const int lane = threadIdx.x & 31;
    const int r    = lane & 15;
    const int half = lane >> 4;

    const float* arow = A + (size_t)(m0 + r) * K + 2 * half;   // frag k-step at arow[k0]
    const float* bcol = Bm + n0 + r;

    v8f acc0 = {}, acc1 = {}, acc2 = {}, acc3 = {};

    // prologue: fragments for k0 = 0
    v2f a_c = *(const v2f*)(arow);
    const float* b0p = bcol + (size_t)(2 * half) * N;
    const float* b1p = b0p + N;
    float bx0 = b0p[0],  by0 = b1p[0];
    float bx1 = b0p[16], by1 = b1p[16];
    float bx2 = b0p[32], by2 = b1p[32];
    float bx3 = b0p[48], by3 = b1p[48];

    // steady state: unconditional next-step loads, then current-step WMMAs
#pragma unroll 4
    for (int k0 = 0; k0 < K - 4; k0 += 4) {
        const v2f a_n = *(const v2f*)(arow + k0 + 4);
        const float* n0p = bcol + (size_t)(k0 + 4 + 2 * half) * N;
        const float* n1p = n0p + N;
        const float nx0 = n0p[0],  ny0 = n1p[0];
        const float nx1 = n0p[16], ny1 = n1p[16];
        const float nx2 = n0p[32], ny2 = n1p[32];
        const float nx3 = n0p[48], ny3 = n1p[48];

        v2f b;
        b.x = bx0; b.y = by0; acc0 = WMMA_F32(a_c, b, acc0);
        b.x = bx1; b.y = by1; acc1 = WMMA_F32(a_c, b, acc1);
        b.x = bx2; b.y = by2; acc2 = WMMA_F32(a_c, b, acc2);
        b.x = bx3; b.y = by3; acc3 = WMMA_F32(a_c, b, acc3);

        a_c = a_n;
        bx0 = nx0; by0 = ny0; bx1 = nx1; by1 = ny1;
        bx2 = nx2; by2 = ny2; bx3 = nx3; by3 = ny3;
    }
    // epilogue: last k-step
    {
        v2f b;
        b.x = bx0; b.y = by0; acc0 = WMMA_F32(a_c, b, acc0);
        b.x = bx1; b.y = by1; acc1 = WMMA_F32(a_c, b, acc1);
        b.x = bx2; b.y = by2; acc2 = WMMA_F32(a_c, b, acc2);
        b.x = bx3; b.y = by3; acc3 = WMMA_F32(a_c, b, acc3);
    }

#pragma unroll
    for (int t = 0; t < 4; ++t) {
        const v8f acc = (t == 0) ? acc0 : (t == 1) ? acc1 : (t == 2) ? acc2 : acc3;
        const int col = n0 + t * 16 + r;
        const float bv = bias[col];
#pragma unroll
        for (int i = 0; i < 8; ++i) {
            const int m = m0 + i + 8 * half;
            C[(size_t)m * N + col] = acc[i] + bv;
        }
    }
}

// ---------------------------------------------------------------------------
// Per-token SE(3) maps of q/k/v 4-chunks, scattering into head-major layout.
//   qx = invP^T per chunk  -> qx[0:3] = R u,        qx3 = it.u[0:3] + u3
//   kx,vx = P per chunk    -> kx[0:3] = R u + t u3, kx3 = u3
// ---------------------------------------------------------------------------
__global__ __launch_bounds__(128) void pose_qkv_kernel(const float* __restrict__ qkv,
                                                       const float* __restrict__ poses,
                                                       float* __restrict__ qx,
                                                       float* __restrict__ kx,
                                                       float* __restrict__ vx) {
    const int token = blockIdx.x;          // b*N + n
    const int c     = threadIdx.x;         // 0..127
    const int h     = c >> 4;
    const int cc    = c & 15;
    const int b     = token / SEQN;
    const int n     = token % SEQN;

    const float* P = poses + (size_t)token * 16;
    const float R00 = P[0],  R01 = P[1],  R02 = P[2],  t0 = P[3];
    const float R10 = P[4],  R11 = P[5],  R12 = P[6],  t1 = P[7];
    const float R20 = P[8],  R21 = P[9],  R22 = P[10], t2 = P[11];
    const float it0 = -(R00 * t0 + R10 * t1 + R20 * t2);
    const float it1 = -(R01 * t0 + R11 * t1 + R21 * t2);
    const float it2 = -(R02 * t0 + R12 * t1 + R22 * t2);

    const float* base = qkv + (size_t)token * QKVW;
    const size_t ho = (((size_t)b * HEADS + h) * SEQN + n) * DHEAD + cc * 4;

    {   // q: invP^T
        const float* u = base + c * 4;
        qx[ho + 0] = R00 * u[0] + R01 * u[1] + R02 * u[2];
        qx[ho + 1] = R10 * u[0] + R11 * u[1] + R12 * u[2];
        qx[ho + 2] = R20 * u[0] + R21 * u[1] + R22 * u[2];
        qx[ho + 3] = it0 * u[0] + it1 * u[1] + it2 * u[2] + u[3];
    }
    {   // k: P
        const float* u = base + INNER + c * 4;
        kx[ho + 0] = R00 * u[0] + R01 * u[1] + R02 * u[2] + t0 * u[3];
        kx[ho + 1] = R10 * u[0] + R11 * u[1] + R12 * u[2] + t1 * u[3];
        kx[ho + 2] = R20 * u[0] + R21 * u[1] + R22 * u[2] + t2 * u[3];
        kx[ho + 3] = u[3];
    }
    {   // v: P
        const float* u = base + 2 * INNER + c * 4;
        vx[ho + 0] = R00 * u[0] + R01 * u[1] + R02 * u[2] + t0 * u[3];
        vx[ho + 1] = R10 * u[0] + R11 * u[1] + R12 * u[2] + t1 * u[3];
        vx[ho + 2] = R20 * u[0] + R21 * u[1] + R22 * u[2] + t2 * u[3];
        vx[ho + 3] = u[3];
    }
}

// ---------------------------------------------------------------------------
// Flash attention, one wave per 16-query tile per (b,h), 64 keys/iteration.
// Q fragment hoisted to registers; K/V/P fragment loads software-pipelined
// one k-step ahead of the consuming WMMAs (guards constant-fold in the
// fully-unrolled loops); P staged via a 16x64 LDS tile.
// ---------------------------------------------------------------------------
__global__ __launch_bounds__(32) void attn_kernel(const float* __restrict__ qx,
                                                  const float* __restrict__ kx,
                                                  const float* __restrict__ vx,
                                                  float* __restrict__ out) {
    __shared__ float ldsP[16 * 64];

    const int q0   = blockIdx.x * 16;
    const int bh   = blockIdx.y;           // b*HEADS + h
    const int lane = threadIdx.x & 31;
    const int r    = lane & 15;
    const int half = lane >> 4;
    const float scale = 0.125f;            // DHEAD^-0.5

    const float* qh = qx + (size_t)bh * SEQN * DHEAD;
    const float* kh = kx + (size_t)bh * SEQN * DHEAD;
    const float* vh = vx + (size_t)bh * SEQN * DHEAD;

    // hoist Q fragments: 16 k-steps x float2 per lane
    v2f qa[16];
    {
        const float* arow = qh + (size_t)(q0 + r) * DHEAD + 2 * half;
#pragma unroll
        for (int kk = 0; kk < 16; ++kk)
            qa[kk] = *(const v2f*)(arow + 4 * kk);
    }

    v8f o0 = {}, o1 = {}, o2 = {}, o3 = {};
    float mrow[8], lrow[8];
#pragma unroll
    for (int i = 0; i < 8; ++i) { mrow[i] = -3.0e38f; lrow[i] = 0.f; }

    for (int kt = 0; kt < SEQN; kt += 64) {
        // prefetch next 64-key block of K (global_prefetch_b8)
        if (kt + 64 < SEQN)
            __builtin_prefetch(kh + (size_t)(kt + 64 + r) * DHEAD, 0, 3);

        // ---- S[4] = Q K^T, four 16x16 tiles, K-dim = 64, pipelined ----
        v8f s0 = {}, s1 = {}, s2 = {}, s3 = {};
        const float* br0 = kh + (size_t)(kt + r) * DHEAD + 2 * half;
        const float* br1 = br0 + 16 * DHEAD;
        const float* br2 = br0 + 32 * DHEAD;
        const float* br3 = br0 + 48 * DHEAD;

        v2f kb0 = *(const v2f*)(br0);
        v2f kb1 = *(const v2f*)(br1);
        v2f kb2 = *(const v2f*)(br2);
        v2f kb3 = *(const v2f*)(br3);
#pragma unroll
        for (int kk = 0; kk < 16; ++kk) {
            v2f nb0 = {}, nb1 = {}, nb2 = {}, nb3 = {};
            if (kk < 15) {                 // constant-folds per unrolled copy
                const int off = 4 * (kk + 1);
                nb0 = *(const v2f*)(br0 + off);
                nb1 = *(const v2f*)(br1 + off);
                nb2 = *(const v2f*)(br2 + off);
                nb3 = *(const v2f*)(br3 + off);
            }
            const v2f a = qa[kk];
            s0 = WMMA_F32(a, kb0, s0);
            s1 = WMMA_F32(a, kb1, s1);
            s2 = WMMA_F32(a, kb2, s2);
            s3 = WMMA_F32(a, kb3, s3);
            kb0 = nb0; kb1 = nb1; kb2 = nb2; kb3 = nb3;
        }

        // ---- online softmax over 64 keys (one shuffle tree per row) ----
        float p0[8], p1[8], p2[8], p3[8], corr[8];
#pragma unroll
        for (int i = 0; i < 8; ++i) {
            const float a0 = s0[i] * scale;
            const float a1 = s1[i] * scale;
            const float a2 = s2[i] * scale;
            const float a3 = s3[i] * scale;
            float rmax = fmaxf(fmaxf(a0, a1), fmaxf(a2, a3));
            for (int m = 1; m < 16; m <<= 1)
                rmax = fmaxf(rmax, __shfl_xor(rmax, m, 32));
            const float mnew = fmaxf(mrow[i], rmax);
            corr[i] = __expf(mrow[i] - mnew);
            p0[i] = __expf(a0 - mnew);
            p1[i] = __expf(a1 - mnew);
            p2[i] = __expf(a2 - mnew);
            p3[i] = __expf(a3 - mnew);
            float rsum = (p0[i] + p1[i]) + (p2[i] + p3[i]);
            for (int m = 1; m < 16; m <<= 1)
                rsum += __shfl_xor(rsum, m, 32);
            lrow[i] = lrow[i] * corr[i] + rsum;
            mrow[i] = mnew;
        }
#pragma unroll
        for (int i = 0; i < 8; ++i) {
            o0[i] *= corr[i]; o1[i] *= corr[i];
            o2[i] *= corr[i]; o3[i] *= corr[i];
        }

        // ---- stage P (16 rows x 64 keys) to LDS ----
        __syncthreads();
#pragma unroll
        for (int i = 0; i < 8; ++i) {
            const int row = i + 8 * half;
            ldsP[row * 64 +  0 + r] = p0[i];
            ldsP[row * 64 + 16 + r] = p1[i];
            ldsP[row * 64 + 32 + r] = p2[i];
            ldsP[row * 64 + 48 + r] = p3[i];
        }
        __syncthreads();

        // ---- O += P @ V (K-dim = 64, N-dim = 64 in 4 tiles), pipelined ----
        const float* lp = ldsP + r * 64 + 2 * half;          // A-frag k-step at lp[4*kk]
        const float* vb = vh + ((size_t)kt + 2 * half) * DHEAD + r;  // b.x row base

        v2f pa = *(const v2f*)(lp);
        {
            const float* v0 = vb;
            const float* v1 = vb + DHEAD;
            float vx0 = v0[0],  vy0 = v1[0];
            float vx1 = v0[16], vy1 = v1[16];
            float vx2 = v0[32], vy2 = v1[32];
            float vx3 = v0[48], vy3 = v1[48];
#pragma unroll
            for (int kk = 0; kk < 16; ++kk) {
                v2f pn = {};
                float mx0 = 0, my0 = 0, mx1 = 0, my1 = 0;
                float mx2 = 0, my2 = 0, mx3 = 0, my3 = 0;
                if (kk < 15) {             // constant-folds per unrolled copy
                    pn = *(const v2f*)(lp + 4 * (kk + 1));
                    const float* n0p = vb + (size_t)(4 * (kk + 1)) * DHEAD;
                    const float* n1p = n0p + DHEAD;
                    mx0 = n0p[0];  my0 = n1p[0];
                    mx1 = n0p[16]; my1 = n1p[16];
                    mx2 = n0p[32]; my2 = n1p[32];
                    mx3 = n0p[48]; my3 = n1p[48];
                }
                v2f b;
                b.x = vx0; b.y = vy0; o0 = WMMA_F32(pa, b, o0);
                b.x = vx1; b.y = vy1; o1 = WMMA_F32(pa, b, o1);
                b.x = vx2; b.y = vy2; o2 = WMMA_F32(pa, b, o2);
                b.x = vx3; b.y = vy3; o3 = WMMA_F32(pa, b, o3);
                pa = pn;
                vx0 = mx0; vy0 = my0; vx1 = mx1; vy1 = my1;
                vx2 = mx2; vy2 = my2; vx3 = mx3; vy3 = my3;
            }
        }
    }

    // ---- normalize + store (head-major) ----
#pragma unroll
    for (int i = 0; i < 8; ++i) {
        const float inv = 1.0f / lrow[i];
        const int m = q0 + i + 8 * half;
        float* orow = out + ((size_t)bh * SEQN + m) * DHEAD;
        orow[ 0 + r] = o0[i] * inv;
        orow[16 + r] = o1[i] * inv;
        orow[32 + r] = o2[i] * inv;
        orow[48 + r] = o3[i] * inv;
    }
}

// ---------------------------------------------------------------------------
// Apply inv-pose rel_map to attention output, fold heads back to [B,N,INNER].
//   o[0:3] = R^T u[0:3] + it * u3 ;  o3 = u3
// ---------------------------------------------------------------------------
__global__ __launch_bounds__(128) void pose_out_kernel(const float* __restrict__ att,
                                                       const float* __restrict__ poses,
                                                       float* __restrict__ mapped) {
    const int token = blockIdx.x;
    const int c     = threadIdx.x;
    const int h     = c >> 4;
    const int cc    = c & 15;
    const int b     = token / SEQN;
    const int n     = token % SEQN;

    const float* P = poses + (size_t)token * 16;
    const float R00 = P[0],  R01 = P[1],  R02 = P[2],  t0 = P[3];
    const float R10 = P[4],  R11 = P[5],  R12 = P[6],  t1 = P[7];
    const float R20 = P[8],  R21 = P[9],  R22 = P[10], t2 = P[11];
    const float it0 = -(R00 * t0 + R10 * t1 + R20 * t2);
    const float it1 = -(R01 * t0 + R11 * t1 + R21 * t2);
    const float it2 = -(R02 * t0 + R12 * t1 + R22 * t2);

    const float* u = att + (((size_t)b * HEADS + h) * SEQN + n) * DHEAD + cc * 4;
    float* o = mapped + (size_t)token * INNER + c * 4;
    o[0] = R00 * u[0] + R10 * u[1] + R20 * u[2] + it0 * u[3];
    o[1] = R01 * u[0] + R11 * u[1] + R21 * u[2] + it1 * u[3];
    o[2] = R02 * u[0] + R12 * u[1] + R22 * u[2] + it2 * u[3];
    o[3] = u[3];
}

// ---------------------------------------------------------------------------
// Host-side launch
// ---------------------------------------------------------------------------
extern "C" void kernel_launch(void* const* d_in, const int* in_sizes, int n_in,
                              void* d_out, int out_size, void* d_ws, size_t ws_size,
                              hipStream_t stream) {
    const float* x      = (const float*)d_in[0];   // [B,N,512]
    const float* poses  = (const float*)d_in[1];   // [B,N,4,4]
    const float* w_qkv  = (const float*)d_in[2];   // [512,1536]
    const float* b_qkv  = (const float*)d_in[3];   // [1536]
    const float* w_out  = (const float*)d_in[4];   // [512,512]
    const float* b_out  = (const float*)d_in[5];   // [512]
    float* out = (float*)d_out;                    // [B,N,512]

    float* ws   = (float*)d_ws;
    float* qkv  = ws;                                      // TOKENS*QKVW floats
    float* qxp  = qkv + (size_t)TOKENS * QKVW;
    float* kxp  = qxp + (size_t)TOKENS * INNER;
    float* vxp  = kxp + (size_t)TOKENS * INNER;
    float* att  = vxp + (size_t)TOKENS * INNER;
    float* mapped = qkv;                                   // reuse qkv region

    // 1) QKV projection: [4096,512] @ [512,1536] + bias
    gemm_bias_wmma<<<dim3(QKVW / 64, TOKENS / 16), 32, 0, stream>>>(
        x, w_qkv, b_qkv, qkv, TOKENS, QKVW, DIM);

    // 2) SE(3) maps into head-major q/k/v
    pose_qkv_kernel<<<TOKENS, 128, 0, stream>>>(qkv, poses, qxp, kxp, vxp);

    // 3) flash attention per (b,h) x 16-query tile
    attn_kernel<<<dim3(SEQN / 16, BATCH * HEADS), 32, 0, stream>>>(qxp, kxp, vxp, att);

    // 4) inverse pose map + head fold
    pose_out_kernel<<<TOKENS, 128, 0, stream>>>(att, poses, mapped);

    // 5) output projection: [4096,512] @ [512,512] + bias
    gemm_bias_wmma<<<dim3(DIM / 64, TOKENS / 16), 32, 0, stream>>>(
        mapped, w_out, b_out, out, TOKENS, DIM, INNER);
}